// SetAbstraction_25177098289642
// MI455X (gfx1250) — compile-verified
//
#include <hip/hip_runtime.h>
#include <hip/hip_bf16.h>
#include <cstddef>

typedef __attribute__((ext_vector_type(2))) float v2f;
typedef __attribute__((ext_vector_type(8))) float v8f;

#define NPTS   8192
#define NCTR   2048
#define NB     8
#define KNBR   32
#define NFEAT  64
#define RADIUS 0.2f

// ---------------------------------------------------------------------------
// Kernel 1: farthest point sampling. One block per batch; 8 points/thread
// cached in registers; per-step argmax via wave shuffle + LDS tree.
// ---------------------------------------------------------------------------
__global__ __launch_bounds__(1024)
void fps_kernel(const float* __restrict__ xyz, float* __restrict__ centers) {
    const int b   = blockIdx.x;
    const int tid = threadIdx.x;
    const float* base = xyz + (size_t)b * NPTS * 3;

    float px[8], py[8], pz[8], mind[8];
    #pragma unroll
    for (int j = 0; j < 8; ++j) {
        int p = j * 1024 + tid;
        px[j] = base[p * 3 + 0];
        py[j] = base[p * 3 + 1];
        pz[j] = base[p * 3 + 2];
        mind[j] = 1e10f;
    }

    __shared__ float rv[32];
    __shared__ int   ri[32];
    __shared__ float lastP[3];
    __shared__ int   bestIdx;

    float lx = base[0], ly = base[1], lz = base[2];
    if (tid == 0) {
        float* c = centers + (size_t)b * NCTR * 3;
        c[0] = lx; c[1] = ly; c[2] = lz;
    }

    for (int i = 1; i < NCTR; ++i) {
        float bd = -1.0f; int bp = 0;
        #pragma unroll
        for (int j = 0; j < 8; ++j) {
            float dx = px[j] - lx, dy = py[j] - ly, dz = pz[j] - lz;
            float d  = dx * dx + dy * dy + dz * dz;
            float m  = fminf(mind[j], d);
            mind[j]  = m;
            int p = j * 1024 + tid;
            if (m > bd || (m == bd && p < bp)) { bd = m; bp = p; }
        }
        // wave32 reduction (argmax, tie -> smallest index)
        #pragma unroll
        for (int off = 16; off > 0; off >>= 1) {
            float od = __shfl_xor(bd, off);
            int   op = __shfl_xor(bp, off);
            if (od > bd || (od == bd && op < bp)) { bd = od; bp = op; }
        }
        int lane = tid & 31, wid = tid >> 5;
        if (lane == 0) { rv[wid] = bd; ri[wid] = bp; }
        __syncthreads();
        if (tid < 32) {
            bd = rv[tid]; bp = ri[tid];
            #pragma unroll
            for (int off = 16; off > 0; off >>= 1) {
                float od = __shfl_xor(bd, off);
                int   op = __shfl_xor(bp, off);
                if (od > bd || (od == bd && op < bp)) { bd = od; bp = op; }
            }
            if (tid == 0) bestIdx = bp;
        }
        __syncthreads();
        int bi = bestIdx;
        if (tid == (bi & 1023)) {
            int oj = bi >> 10;
            lastP[0] = px[oj]; lastP[1] = py[oj]; lastP[2] = pz[oj];
            float* c = centers + ((size_t)b * NCTR + i) * 3;
            c[0] = px[oj]; c[1] = py[oj]; c[2] = pz[oj];
        }
        __syncthreads();
        lx = lastP[0]; ly = lastP[1]; lz = lastP[2];
    }
}

// ---------------------------------------------------------------------------
// Kernel 2: ball query. One wave32 per center; first-K-by-index selection via
// ballot + prefix popcount, early exit once K slots are filled.
// ---------------------------------------------------------------------------
__global__ __launch_bounds__(256)
void ballquery_kernel(const float* __restrict__ centers,
                      const float* __restrict__ xyz,
                      int* __restrict__ gidx) {
    int w    = (int)((blockIdx.x * blockDim.x + threadIdx.x) >> 5);
    int lane = threadIdx.x & 31;
    if (w >= NB * NCTR) return;
    int b = w >> 11;

    const float* c = centers + (size_t)w * 3;
    float cx = c[0], cy = c[1], cz = c[2];
    const float* base = xyz + (size_t)b * NPTS * 3;
    const float r2 = RADIUS * RADIUS;

    int* og = gidx + (size_t)w * KNBR;
    int filled = 0;
    for (int chunk = 0; chunk < NPTS / 32; ++chunk) {
        int p = chunk * 32 + lane;
        float dx = base[p * 3 + 0] - cx;
        float dy = base[p * 3 + 1] - cy;
        float dz = base[p * 3 + 2] - cz;
        bool within = (dx * dx + dy * dy + dz * dz) <= r2;
        unsigned mask = __builtin_amdgcn_ballot_w32(within);
        if (within) {
            int slot = filled + __builtin_popcount(mask & ((1u << lane) - 1u));
            if (slot < KNBR) og[slot] = p;
        }
        filled += __builtin_popcount(mask);
        if (filled >= KNBR) break;
    }
    if (filled < KNBR && lane >= filled) og[lane] = -1;
}

// ---------------------------------------------------------------------------
// fp32 WMMA tile: acc(16x16) += X(16x4) * W^T(4x16), looping K in steps of 4.
// A layout: lane&15 = M row, lane>=16 supplies K+2/K+3.
// B layout: lane&15 = N col (mirrored), element = W[n][k].
// ---------------------------------------------------------------------------
__device__ __forceinline__
v8f tile_gemm(const float* __restrict__ X, int sx,
              const float* __restrict__ W, int sw,
              int kSteps, int mrow, int ncol, int khalf, v8f acc) {
    for (int ks = 0; ks < kSteps; ++ks) {
        int k = ks * 4 + khalf;
        v2f a;  a.x  = X[mrow * sx + k];  a.y  = X[mrow * sx + k + 1];
        v2f bb; bb.x = W[ncol * sw + k];  bb.y = W[ncol * sw + k + 1];
        acc = __builtin_amdgcn_wmma_f32_16x16x4_f32(
            false, a, false, bb, (short)0, acc, false, false);
    }
    return acc;
}

// ---------------------------------------------------------------------------
// Kernel 3: gather + 3-layer MLP (fused BN scale/shift + relu) + max over K.
// 128 threads = 4 waves cooperate on one center; wave w owns N-tile(s) w.
// ---------------------------------------------------------------------------
__global__ __launch_bounds__(128)
void sa_mlp_kernel(const float* __restrict__ xyz, const float* __restrict__ feats,
                   const float* __restrict__ centers, const int* __restrict__ gidx,
                   const float* __restrict__ w1, const float* __restrict__ b1,
                   const float* __restrict__ g1, const float* __restrict__ bt1,
                   const float* __restrict__ m1, const float* __restrict__ v1,
                   const float* __restrict__ w2, const float* __restrict__ b2,
                   const float* __restrict__ g2, const float* __restrict__ bt2,
                   const float* __restrict__ m2, const float* __restrict__ v2,
                   const float* __restrict__ w3, const float* __restrict__ b3,
                   const float* __restrict__ g3, const float* __restrict__ bt3,
                   const float* __restrict__ m3, const float* __restrict__ v3,
                   float* __restrict__ outfeat) {
    __shared__ float W1s[64 * 68];   // padded 67 -> 68 (zero pad col)
    __shared__ float W2s[64 * 64];
    __shared__ float Xa[32 * 68];    // activations ping
    __shared__ float Xb[32 * 64];    // activations pong
    __shared__ float sc1[64], sh1[64], sc2[64], sh2[64], sc3[128], sh3[128];

    const int tid  = threadIdx.x;
    const int lane = tid & 31;
    const int wid  = tid >> 5;

    for (int i = tid; i < 64 * 68; i += 128) {
        int n = i / 68, k = i - n * 68;
        W1s[i] = (k < 67) ? w1[n * 67 + k] : 0.0f;
    }
    for (int i = tid; i < 64 * 64; i += 128) W2s[i] = w2[i];
    if (tid < 64) {
        float s = g1[tid] * rsqrtf(v1[tid] + 1e-5f);
        sc1[tid] = s; sh1[tid] = (b1[tid] - m1[tid]) * s + bt1[tid];
        s = g2[tid] * rsqrtf(v2[tid] + 1e-5f);
        sc2[tid] = s; sh2[tid] = (b2[tid] - m2[tid]) * s + bt2[tid];
    }
    {
        float s = g3[tid] * rsqrtf(v3[tid] + 1e-5f);
        sc3[tid] = s; sh3[tid] = (b3[tid] - m3[tid]) * s + bt3[tid];
    }
    __syncthreads();

    const int khalf = (lane >> 4) * 2;
    const int mloc  = lane & 15;

    for (int c = blockIdx.x; c < NB * NCTR; c += gridDim.x) {
        int b = c >> 11;

        // ---- gather: row = tid>>2 (K=32 rows), 4 threads per row split cols
        {
            int row = tid >> 2, q = tid & 3;
            int gi = gidx[(size_t)c * KNBR + row];
            bool valid = gi >= 0;
            int j = valid ? gi : 0;
            const float* p   = xyz   + ((size_t)b * NPTS + j) * 3;
            const float* ctr = centers + (size_t)c * 3;
            const float* fr  = feats + ((size_t)b * NPTS + j) * NFEAT;
            #pragma unroll
            for (int t = 0; t < 17; ++t) {
                int col = q * 17 + t;
                float val;
                if (col < 3)       val = valid ? (p[col] - ctr[col]) : 0.0f;
                else if (col < 67) val = valid ? fr[col - 3] : 0.0f;
                else               val = 0.0f;
                Xa[row * 68 + col] = val;
            }
        }
        __syncthreads();

        // ---- layer 1: Xa(32x68) x W1^T -> Xb(32x64); wave wid owns nt = wid
        {
            int ncol = wid * 16 + mloc;
            float scl = sc1[ncol], sft = sh1[ncol];
            #pragma unroll
            for (int mt = 0; mt < 2; ++mt) {
                v8f acc = {};
                acc = tile_gemm(Xa, 68, W1s, 68, 17, mt * 16 + mloc, ncol, khalf, acc);
                #pragma unroll
                for (int r = 0; r < 8; ++r) {
                    float y = fmaxf(acc[r] * scl + sft, 0.0f);
                    int orow = mt * 16 + r + ((lane >> 4) << 3);
                    Xb[orow * 64 + ncol] = y;
                }
            }
        }
        __syncthreads();

        // ---- layer 2: Xb(32x64) x W2^T -> Xa (stride 64)
        {
            int ncol = wid * 16 + mloc;
            float scl = sc2[ncol], sft = sh2[ncol];
            #pragma unroll
            for (int mt = 0; mt < 2; ++mt) {
                v8f acc = {};
                acc = tile_gemm(Xb, 64, W2s, 64, 16, mt * 16 + mloc, ncol, khalf, acc);
                #pragma unroll
                for (int r = 0; r < 8; ++r) {
                    float y = fmaxf(acc[r] * scl + sft, 0.0f);
                    int orow = mt * 16 + r + ((lane >> 4) << 3);
                    Xa[orow * 64 + ncol] = y;
                }
            }
        }
        __syncthreads();

        // ---- layer 3: Xa(32x64) x W3^T(global) -> max over 32 rows -> out
        {
            #pragma unroll
            for (int t = 0; t < 2; ++t) {
                int nt = wid * 2 + t;
                int ncol = nt * 16 + mloc;
                float scl = sc3[ncol], sft = sh3[ncol];
                float cm = 0.0f;                      // relu outputs are >= 0
                #pragma unroll
                for (int mt = 0; mt < 2; ++mt) {
                    v8f acc = {};
                    acc = tile_gemm(Xa, 64, w3, 64, 16, mt * 16 + mloc, ncol, khalf, acc);
                    float pm = 0.0f;
                    #pragma unroll
                    for (int r = 0; r < 8; ++r)
                        pm = fmaxf(pm, fmaxf(acc[r] * scl + sft, 0.0f));
                    pm = fmaxf(pm, __shfl_xor(pm, 16));  // merge row-halves
                    cm = fmaxf(cm, pm);
                }
                if (lane < 16)
                    outfeat[(size_t)c * 128 + ncol] = cm;
            }
        }
        __syncthreads();
    }
}

// ---------------------------------------------------------------------------
extern "C" void kernel_launch(void* const* d_in, const int* in_sizes, int n_in,
                              void* d_out, int out_size, void* d_ws, size_t ws_size,
                              hipStream_t stream) {
    const float* xyzs  = (const float*)d_in[0];
    const float* feats = (const float*)d_in[1];
    const float* w1 = (const float*)d_in[2],  *b1 = (const float*)d_in[3];
    const float* g1 = (const float*)d_in[4],  *bt1 = (const float*)d_in[5];
    const float* m1 = (const float*)d_in[6],  *v1 = (const float*)d_in[7];
    const float* w2 = (const float*)d_in[8],  *b2 = (const float*)d_in[9];
    const float* g2 = (const float*)d_in[10], *bt2 = (const float*)d_in[11];
    const float* m2 = (const float*)d_in[12], *v2 = (const float*)d_in[13];
    const float* w3 = (const float*)d_in[14], *b3 = (const float*)d_in[15];
    const float* g3 = (const float*)d_in[16], *bt3 = (const float*)d_in[17];
    const float* m3 = (const float*)d_in[18], *v3 = (const float*)d_in[19];

    float* out     = (float*)d_out;
    float* centers = out;                         // B*S*3 = 49152 floats
    float* outfeat = out + (size_t)NB * NCTR * 3; // B*S*128
    int*   gidx    = (int*)d_ws;                  // B*S*K ints = 2 MB

    fps_kernel<<<dim3(NB), dim3(1024), 0, stream>>>(xyzs, centers);
    ballquery_kernel<<<dim3((NB * NCTR) / 8), dim3(256), 0, stream>>>(centers, xyzs, gidx);
    sa_mlp_kernel<<<dim3(2048), dim3(128), 0, stream>>>(
        xyzs, feats, centers, gidx,
        w1, b1, g1, bt1, m1, v1,
        w2, b2, g2, bt2, m2, v2,
        w3, b3, g3, bt3, m3, v3,
        outfeat);
}